// AdditiveAttention_16312285790458
// MI455X (gfx1250) — compile-verified
//
#include <hip/hip_runtime.h>
#include <hip/hip_bf16.h>

// Sizes fixed by the reference: B=16, NQ=NK=256, D=H=V=256.
#define BATCH   16
#define NQ      256
#define NK      256
#define HDIM    256
#define MASKV   (-1000000.0f)

#define CHW     16                     // h-chunk width (fp32 elements)
#define CHP     17                     // padded LDS row stride (TDM pad: 16+1 dwords)
#define NCH     (HDIM / CHW)           // 16 chunks
#define CHUNK_BYTES (NK * CHP * 4)     // 17,408 B per buffer

typedef float v2f  __attribute__((ext_vector_type(2)));
typedef float v8f  __attribute__((ext_vector_type(8)));
typedef unsigned int u32x4 __attribute__((ext_vector_type(4)));
typedef int   i32x4 __attribute__((ext_vector_type(4)));
typedef int   i32x8 __attribute__((ext_vector_type(8)));

#if defined(__has_builtin)
#if __has_builtin(__builtin_amdgcn_tensor_load_to_lds) && __has_builtin(__builtin_amdgcn_s_wait_tensorcnt)
#define USE_TDM 1
#endif
#endif

// ---------------------------------------------------------------------------
// fp32 hardware tanh (gfx1250 TRANS op). Builtin if present, else inline asm
// with a trailing v_nop to satisfy the 1-op TRANS hazard rule.
// ---------------------------------------------------------------------------
__device__ __forceinline__ float fast_tanh(float x) {
#if defined(__has_builtin) && __has_builtin(__builtin_amdgcn_tanhf)
  return __builtin_amdgcn_tanhf(x);
#else
  float r;
  asm volatile("v_tanh_f32 %0, %1\n\tv_nop" : "=v"(r) : "v"(x));
  return r;
#endif
}

#ifdef USE_TDM
// ---------------------------------------------------------------------------
// TDM: DMA one [256 rows x 16 f32] tile (row stride 256 f32 in global) into
// LDS at lds_off, with 1-dword padding after every 16 dwords -> LDS row
// stride 17 (bank-conflict-free for the per-thread consumers).
// D# layout per ISA 8.3/8.4 (group0 128b, group1 256b; groups 2/3 zero = 2D).
// This toolchain exposes the 6-arg builtin:
//   (u32x4 g0, i32x8 g1, i32x4 g2, i32x4 g3, i32x8 extra, i32 cpol)
// ---------------------------------------------------------------------------
__device__ __forceinline__ void tdm_load_chunk(const float* gptr, unsigned lds_off) {
  unsigned long long ga = (unsigned long long)(uintptr_t)gptr;
  u32x4 g0;
  g0[0] = 1u;                                              // count=1, no gather
  g0[1] = lds_off;                                         // lds_addr
  g0[2] = (unsigned)(ga & 0xFFFFFFFFu);                    // global_addr[31:0]
  g0[3] = (unsigned)((ga >> 32) & 0x1FFFFFFu) | (2u << 30); // addr[56:32] | type=2
  i32x8 g1;
  g1[0] = (2 << 16)        // data_size = 4B
        | (1 << 20)        // pad_enable
        | (3 << 22)        // pad_interval: 16 dwords
        | (0 << 25);       // pad_amount: 1 dword
  g1[1] = (CHW & 0xFFFF) << 16;            // tensor_dim0[15:0] = 16
  g1[2] = (NK  & 0xFFFF) << 16;            // tensor_dim0 hi=0 | tensor_dim1[15:0]=256
  g1[3] = (CHW & 0xFFFF) << 16;            // tensor_dim1 hi=0 | tile_dim0=16
  g1[4] = NK;                              // tile_dim1=256 | tile_dim2=0
  g1[5] = HDIM;                            // tensor_dim0_stride lo32 = 256 elems
  g1[6] = 0;                               // stride hi | tensor_dim1_stride lo
  g1[7] = 0;                               // tensor_dim1_stride hi
  i32x4 z4 = {0, 0, 0, 0};
  i32x8 z8 = {0, 0, 0, 0, 0, 0, 0, 0};
  __builtin_amdgcn_tensor_load_to_lds(g0, g1, z4, z4, z8, 0);
}
#endif

// ---------------------------------------------------------------------------
// Generic 256-column GEMM: C[z] = A[z] (Mx256) * B[z] (256x256), fp32 WMMA.
// Block = 256 threads = 8 waves; block tile 64(M) x 64(N).
// Wave (wm in 0..3, wn in 0..1) owns a 16x32 strip: 2 accumulators,
// A fragment reused across both WMMAs. K loop: 64 steps of 16x16x4.
// ---------------------------------------------------------------------------
__global__ __launch_bounds__(256) void gemm256_wmma(
    const float* __restrict__ A, const float* __restrict__ Bm,
    float* __restrict__ C, int aStride, int bStride, int cStride)
{
  const int z = blockIdx.z;
  A  += (long)z * aStride;
  Bm += (long)z * bStride;
  C  += (long)z * cStride;

  const int lane = threadIdx.x & 31;
  const int wave = threadIdx.x >> 5;
  const int wm   = wave & 3;
  const int wn   = wave >> 2;
  const int l15  = lane & 15;
  const int hi   = lane >> 4;

  const int rowBase = blockIdx.x * 64 + wm * 16;
  const int colBase = blockIdx.y * 64 + wn * 32;

  const float* aRow = A  + (rowBase + l15) * HDIM + hi * 2;
  const float* b0p  = Bm + (hi * 2) * HDIM + colBase + l15;
  const float* b1p  = b0p + 16;

  v8f acc0 = {};
  v8f acc1 = {};

#pragma unroll 8
  for (int k = 0; k < HDIM; k += 4) {
    v2f a = *(const v2f*)(aRow + k);
    v2f b0, b1;
    b0.x = b0p[(k)     * HDIM];
    b0.y = b0p[(k + 1) * HDIM];
    b1.x = b1p[(k)     * HDIM];
    b1.y = b1p[(k + 1) * HDIM];
    acc0 = __builtin_amdgcn_wmma_f32_16x16x4_f32(false, a, false, b0,
                                                 (short)0, acc0, false, false);
    acc1 = __builtin_amdgcn_wmma_f32_16x16x4_f32(false, a, false, b1,
                                                 (short)0, acc1, false, false);
  }

  const int row0 = rowBase + hi * 8;
  const int col  = colBase + l15;
#pragma unroll
  for (int r = 0; r < 8; ++r) {
    C[(row0 + r) * HDIM + col]      = acc0[r];
    C[(row0 + r) * HDIM + col + 16] = acc1[r];
  }
}

// ---------------------------------------------------------------------------
// Fused scores + mask + softmax. One block per (b, q) row; thread t owns key t.
//   score[t] = sum_h w_v[h] * tanh(Qp[b,q,h] + Kp[b,t,h])
// Kp is streamed through double-buffered LDS chunks via the Tensor Data Mover
// (one tensor_load_to_lds per chunk, prefetch overlapped with the tanh loop,
// synchronized with s_wait_tensorcnt + workgroup barrier). Then masked
// block-wide softmax.
// ---------------------------------------------------------------------------
__global__ __launch_bounds__(256) void attn_scores_softmax(
    const float* __restrict__ Qp, const float* __restrict__ Kp,
    const float* __restrict__ w_v, const int* __restrict__ valid_lens,
    float* __restrict__ attn)
{
  const int bq  = blockIdx.x;        // 0 .. B*NQ-1
  const int b   = bq >> 8;           // NQ == 256
  const int tid = threadIdx.x;       // key index

  __shared__ float lds_q[HDIM];
  __shared__ float lds_w[HDIM];
  __shared__ float lds_k[2][NK * CHP];   // double-buffered padded tiles (~35 KB)
  __shared__ float red[256];

  lds_q[tid] = Qp[bq * HDIM + tid];
  lds_w[tid] = w_v[tid];

  const float* KpB = Kp + b * (NK * HDIM);
  float acc = 0.0f;

#ifdef USE_TDM
  const unsigned lds_base = (unsigned)(uintptr_t)(&lds_k[0][0]);
  if (tid < 32) {                                   // wave 0 drives the TDM
    tdm_load_chunk(KpB, lds_base);                  // prologue: chunk 0 -> buf 0
  }
  for (int hc = 0; hc < NCH; ++hc) {
    const int buf = hc & 1;
    if (tid < 32) {
      if (hc + 1 < NCH) {
        tdm_load_chunk(KpB + (hc + 1) * CHW,
                       lds_base + (unsigned)((buf ^ 1) * CHUNK_BYTES));
        __builtin_amdgcn_s_wait_tensorcnt(1);       // chunk hc done, hc+1 in flight
      } else {
        __builtin_amdgcn_s_wait_tensorcnt(0);
      }
    }
    __syncthreads();                                // publish buf to all waves
    const int h0 = hc * CHW;
    const float* kr = &lds_k[buf][tid * CHP];
#pragma unroll
    for (int c = 0; c < CHW; ++c) {
      float x = lds_q[h0 + c] + kr[c];
      acc = fmaf(lds_w[h0 + c], fast_tanh(x), acc);
    }
    __syncthreads();                                // buf free for next DMA
  }
#else
  for (int hc = 0; hc < NCH; ++hc) {
    const int buf = hc & 1;
#pragma unroll
    for (int i = 0; i < CHW; ++i) {                 // cooperative coalesced load
      int f = tid + i * 256;                        // 0 .. 4095
      int r = f >> 4, c = f & (CHW - 1);
      lds_k[buf][r * CHP + c] = KpB[r * HDIM + hc * CHW + c];
    }
    __syncthreads();
    const int h0 = hc * CHW;
    const float* kr = &lds_k[buf][tid * CHP];
#pragma unroll
    for (int c = 0; c < CHW; ++c) {
      float x = lds_q[h0 + c] + kr[c];
      acc = fmaf(lds_w[h0 + c], fast_tanh(x), acc);
    }
    __syncthreads();
  }
#endif

  const int vl = valid_lens[b];
  float score = (tid < vl) ? acc : MASKV;

  // block max
  red[tid] = score;
  __syncthreads();
#pragma unroll
  for (int s = 128; s > 0; s >>= 1) {
    if (tid < s) red[tid] = fmaxf(red[tid], red[tid + s]);
    __syncthreads();
  }
  const float m = red[0];
  __syncthreads();

  const float e = __expf(score - m);
  red[tid] = e;
  __syncthreads();
#pragma unroll
  for (int s = 128; s > 0; s >>= 1) {
    if (tid < s) red[tid] += red[tid + s];
    __syncthreads();
  }
  const float denom = red[0];

  attn[bq * NK + tid] = e / denom;
}

// ---------------------------------------------------------------------------
// Launch: 2 projection GEMMs -> fused score/softmax -> batched attn x V GEMM.
// Workspace: Qp (4MB) | Kp (4MB) | attn (4MB).
// ---------------------------------------------------------------------------
extern "C" void kernel_launch(void* const* d_in, const int* in_sizes, int n_in,
                              void* d_out, int out_size, void* d_ws, size_t ws_size,
                              hipStream_t stream) {
  (void)in_sizes; (void)n_in; (void)out_size; (void)ws_size;

  const float* queries    = (const float*)d_in[0];
  const float* keys       = (const float*)d_in[1];
  const float* values     = (const float*)d_in[2];
  const int*   valid_lens = (const int*)  d_in[3];
  const float* W_q        = (const float*)d_in[4];
  const float* W_k        = (const float*)d_in[5];
  const float* w_v        = (const float*)d_in[6];
  float*       out        = (float*)d_out;

  const int perMat = BATCH * NQ * HDIM;      // 1,048,576 floats
  float* Qp   = (float*)d_ws;
  float* Kp   = Qp + perMat;
  float* attn = Kp + perMat;

  // Qp = queries * W_q   (flattened [B*NQ, 256] x [256, 256])
  gemm256_wmma<<<dim3((BATCH * NQ) / 64, HDIM / 64, 1), 256, 0, stream>>>(
      queries, W_q, Qp, 0, 0, 0);
  // Kp = keys * W_k
  gemm256_wmma<<<dim3((BATCH * NK) / 64, HDIM / 64, 1), 256, 0, stream>>>(
      keys, W_k, Kp, 0, 0, 0);

  // scores + mask + softmax
  attn_scores_softmax<<<dim3(BATCH * NQ, 1, 1), 256, 0, stream>>>(
      Qp, Kp, w_v, valid_lens, attn);

  // out[b] = attn[b] * values[b]   (batched over z)
  gemm256_wmma<<<dim3(NQ / 64, HDIM / 64, BATCH), 256, 0, stream>>>(
      attn, values, out, NQ * NK, NK * HDIM, NQ * HDIM);
}